// ProbTraffic_gat_dgl_pool_25134148616277
// MI455X (gfx1250) — compile-verified
//
#include <hip/hip_runtime.h>
#include <hip/hip_bf16.h>
#include <math.h>

#define N_NODES 100000
#define N_EDGES 1000000
#define IN_DIM  64
#define H0      4
#define D0      32
#define HD0     128      // H0*D0
#define C1      32
#define NEG_SLOPE 0.2f

typedef __attribute__((ext_vector_type(2))) float v2f;
typedef __attribute__((ext_vector_type(8))) float v8f;

// ---------------- fp32 WMMA GEMM: Y[M x OUT] = X[M x K] * W[K x OUT], row-major
// One wave (32 threads) per 16x16 output tile; K-loop in steps of 4 using
// V_WMMA_F32_16X16X4_F32 (exact fp32 math, matches the fp32 reference).
__global__ __launch_bounds__(32) void gat_gemm_f32_wmma(
    const float* __restrict__ X, const float* __restrict__ W,
    float* __restrict__ Y, int M, int K, int OUT) {
  const int tiles_n = OUT >> 4;
  const int tile = blockIdx.x;
  const int tm = tile / tiles_n;
  const int tn = tile % tiles_n;
  const int lane = threadIdx.x;
  const int half = lane >> 4;     // 0: lanes 0-15, 1: lanes 16-31
  const int l15  = lane & 15;
  const int m = (tm << 4) + l15;  // A row held by this lane
  const int n = (tn << 4) + l15;  // B/D column held by this lane
  const int kp = half << 1;       // K sub-offset (0 or 2) within each K=4 step

  v8f acc = {};
  for (int k = 0; k < K; k += 4) {
    v2f a, b;
    // A 16x4 layout: VGPR0/1 = K{kp, kp+1} for this lane's row m
    a.x = X[(size_t)m * K + k + kp];
    a.y = X[(size_t)m * K + k + kp + 1];
    // B 4x16 layout: row K striped across lanes (N = l15), K split by lane half
    b.x = W[(size_t)(k + kp) * OUT + n];
    b.y = W[(size_t)(k + kp + 1) * OUT + n];
    acc = __builtin_amdgcn_wmma_f32_16x16x4_f32(
        /*neg_a=*/false, a, /*neg_b=*/false, b,
        /*c_mod=*/(short)0, acc, /*reuse_a=*/false, /*reuse_b=*/false);
  }
  // C/D layout: VGPR i -> row (i + half*8), col n
  const int rbase = (tm << 4) + (half << 3);
#pragma unroll
  for (int i = 0; i < 8; ++i) {
    Y[(size_t)(rbase + i) * OUT + n] = acc[i];
  }
}

// ---------------- per-node attention dot products: el/er [n_nodes*Hh]
__global__ void gat_node_dots(const float* __restrict__ h,
                              const float* __restrict__ al,
                              const float* __restrict__ ar,
                              float* __restrict__ el, float* __restrict__ er,
                              int n_nodes, int Hh, int Dd) {
  int idx = blockIdx.x * blockDim.x + threadIdx.x;
  if (idx >= n_nodes * Hh) return;
  int n = idx / Hh, hh = idx % Hh;
  const float* hp  = h + (size_t)n * Hh * Dd + hh * Dd;
  const float* alp = al + hh * Dd;
  const float* arp = ar + hh * Dd;
  float sl = 0.f, sr = 0.f;
  for (int d = 0; d < Dd; ++d) { float v = hp[d]; sl += v * alp[d]; sr += v * arp[d]; }
  el[idx] = sl;
  er[idx] = sr;
}

// monotone float <-> uint key for atomicMax-based segment max
__device__ __forceinline__ unsigned fkey(float f) {
  unsigned b = __float_as_uint(f);
  return (b & 0x80000000u) ? ~b : (b | 0x80000000u);
}
__device__ __forceinline__ float funkey(unsigned k) {
  unsigned b = (k & 0x80000000u) ? (k & 0x7FFFFFFFu) : ~k;
  return __uint_as_float(b);
}
__device__ __forceinline__ float lrelu(float x) {
  return x >= 0.f ? x : NEG_SLOPE * x;
}

// ---------------- edge pass 1: segment max of leaky-relu(el[src]+er[dst]) into dst
__global__ void gat_edge_max(const int* __restrict__ src, const int* __restrict__ dst,
                             const float* __restrict__ el, const float* __restrict__ er,
                             unsigned* __restrict__ menc, int nE, int Hh) {
  int e = blockIdx.x * blockDim.x + threadIdx.x;
  if (e >= nE) return;
  int s = src[e], d = dst[e];
  for (int hh = 0; hh < Hh; ++hh) {
    float x = lrelu(el[s * Hh + hh] + er[d * Hh + hh]);
    atomicMax(&menc[d * Hh + hh], fkey(x));
  }
}

// ---------------- edge pass 2: segment sum of exp(e - m[dst])
__global__ void gat_edge_expsum(const int* __restrict__ src, const int* __restrict__ dst,
                                const float* __restrict__ el, const float* __restrict__ er,
                                const unsigned* __restrict__ menc,
                                float* __restrict__ ssum, int nE, int Hh) {
  int e = blockIdx.x * blockDim.x + threadIdx.x;
  if (e >= nE) return;
  int s = src[e], d = dst[e];
  for (int hh = 0; hh < Hh; ++hh) {
    float x = lrelu(el[s * Hh + hh] + er[d * Hh + hh]);
    float ex = expf(x - funkey(menc[d * Hh + hh]));
    atomicAdd(&ssum[d * Hh + hh], ex);
  }
}

// ---------------- edge pass 3: out[dst] += h[src] * alpha
// one block per edge, Hh*Dd threads: fully coalesced gather + L2-resident atomics
__global__ void gat_edge_scatter(const int* __restrict__ src, const int* __restrict__ dst,
                                 const float* __restrict__ h,
                                 const float* __restrict__ el, const float* __restrict__ er,
                                 const unsigned* __restrict__ menc,
                                 const float* __restrict__ ssum,
                                 float* __restrict__ out, int Hh, int Dd) {
  int e = blockIdx.x;
  int t = threadIdx.x;            // 0 .. Hh*Dd-1
  int hh = t / Dd;
  int s = src[e], d = dst[e];
  float x = lrelu(el[s * Hh + hh] + er[d * Hh + hh]);
  float alpha = expf(x - funkey(menc[d * Hh + hh])) / ssum[d * Hh + hh];
  atomicAdd(&out[(size_t)d * Hh * Dd + t], h[(size_t)s * Hh * Dd + t] * alpha);
}

// ---------------- elementwise ELU (in place)
__global__ void gat_elu(float* __restrict__ x, int n) {
  int i = blockIdx.x * blockDim.x + threadIdx.x;
  if (i >= n) return;
  float v = x[i];
  x[i] = v > 0.f ? v : expm1f(v);
}

// ---------------- mean over nodes -> 32 floats
__global__ void gat_mean_pool(const float* __restrict__ x, float* __restrict__ out,
                              int n_nodes) {
  __shared__ float sm[256];
  int t = threadIdx.x;
  int c = t & 31;                 // column 0..31
  int g = t >> 5;                 // row group 0..7
  float s = 0.f;
  for (int n = blockIdx.x * 8 + g; n < n_nodes; n += gridDim.x * 8)
    s += x[(size_t)n * 32 + c];
  sm[t] = s;
  __syncthreads();
  if (t < 128) sm[t] += sm[t + 128];
  __syncthreads();
  if (t < 64) sm[t] += sm[t + 64];
  __syncthreads();
  if (t < 32) atomicAdd(&out[c], (sm[t] + sm[t + 32]) * (1.0f / n_nodes));
}

extern "C" void kernel_launch(void* const* d_in, const int* in_sizes, int n_in,
                              void* d_out, int out_size, void* d_ws, size_t ws_size,
                              hipStream_t stream) {
  const float* T   = (const float*)d_in[0];
  const int*   src = (const int*)  d_in[1];
  const int*   dst = (const int*)  d_in[2];
  const float* W0  = (const float*)d_in[3];
  const float* al0 = (const float*)d_in[4];
  const float* ar0 = (const float*)d_in[5];
  const float* W1  = (const float*)d_in[6];
  const float* al1 = (const float*)d_in[7];
  const float* ar1 = (const float*)d_in[8];
  float* out = (float*)d_out;

  // ---- workspace layout (layer-1 buffers overlay layer-0's dead h0 region)
  float*    h0    = (float*)d_ws;                         // N*HD0
  float*    agg0  = h0 + (size_t)N_NODES * HD0;           // N*HD0 (becomes x1 after ELU)
  float*    el0   = agg0 + (size_t)N_NODES * HD0;         // N*H0
  float*    er0   = el0 + (size_t)N_NODES * H0;           // N*H0
  unsigned* menc0 = (unsigned*)(er0 + (size_t)N_NODES * H0); // N*H0
  float*    ssum0 = (float*)menc0 + (size_t)N_NODES * H0; // N*H0
  float*    h1    = h0;                                   // N*C1 (overlay)
  float*    agg1  = h0 + (size_t)N_NODES * C1;            // N*C1 (overlay)
  float*    el1   = el0;
  float*    er1   = er0;
  unsigned* menc1 = menc0;
  float*    ssum1 = ssum0;

  const int TPB = 256;

  // ================= layer 0 =================
  hipMemsetAsync(agg0,  0, (size_t)N_NODES * HD0 * sizeof(float), stream);
  hipMemsetAsync(menc0, 0, (size_t)N_NODES * H0 * sizeof(unsigned), stream);
  hipMemsetAsync(ssum0, 0, (size_t)N_NODES * H0 * sizeof(float), stream);

  // h0 = T @ W0  : M=100000, K=64, OUT=128 -> 6250*8 wave-tiles
  gat_gemm_f32_wmma<<<(N_NODES / 16) * (HD0 / 16), 32, 0, stream>>>(
      T, W0, h0, N_NODES, IN_DIM, HD0);

  gat_node_dots<<<(N_NODES * H0 + TPB - 1) / TPB, TPB, 0, stream>>>(
      h0, al0, ar0, el0, er0, N_NODES, H0, D0);

  gat_edge_max<<<(N_EDGES + TPB - 1) / TPB, TPB, 0, stream>>>(
      src, dst, el0, er0, menc0, N_EDGES, H0);
  gat_edge_expsum<<<(N_EDGES + TPB - 1) / TPB, TPB, 0, stream>>>(
      src, dst, el0, er0, menc0, ssum0, N_EDGES, H0);
  gat_edge_scatter<<<N_EDGES, HD0, 0, stream>>>(
      src, dst, h0, el0, er0, menc0, ssum0, agg0, H0, D0);

  gat_elu<<<((size_t)N_NODES * HD0 + TPB - 1) / TPB, TPB, 0, stream>>>(
      agg0, N_NODES * HD0);

  // ================= layer 1 =================
  hipMemsetAsync(agg1,  0, (size_t)N_NODES * C1 * sizeof(float), stream);
  hipMemsetAsync(menc1, 0, (size_t)N_NODES * sizeof(unsigned), stream);
  hipMemsetAsync(ssum1, 0, (size_t)N_NODES * sizeof(float), stream);

  // h1 = elu(agg0) @ W1 : M=100000, K=128, OUT=32 -> 6250*2 wave-tiles
  gat_gemm_f32_wmma<<<(N_NODES / 16) * (C1 / 16), 32, 0, stream>>>(
      agg0, W1, h1, N_NODES, HD0, C1);

  gat_node_dots<<<(N_NODES + TPB - 1) / TPB, TPB, 0, stream>>>(
      h1, al1, ar1, el1, er1, N_NODES, 1, C1);

  gat_edge_max<<<(N_EDGES + TPB - 1) / TPB, TPB, 0, stream>>>(
      src, dst, el1, er1, menc1, N_EDGES, 1);
  gat_edge_expsum<<<(N_EDGES + TPB - 1) / TPB, TPB, 0, stream>>>(
      src, dst, el1, er1, menc1, ssum1, N_EDGES, 1);
  gat_edge_scatter<<<N_EDGES, C1, 0, stream>>>(
      src, dst, h1, el1, er1, menc1, ssum1, agg1, 1, C1);

  // ================= mean pool =================
  hipMemsetAsync(out, 0, 32 * sizeof(float), stream);
  gat_mean_pool<<<256, 256, 0, stream>>>(agg1, out, N_NODES);
}